// vect_minimax_disc_Loss_41523743817772
// MI455X (gfx1250) — compile-verified
//
#include <hip/hip_runtime.h>

#define D 256
#define C 100
#define CP 112          // classes padded to 7*16 for WMMA tiling
#define COLS 64         // feature columns handled per workgroup (LDS budget)
#define ROW_GROUPS 256  // row chunks; grid = ROW_GROUPS x (D/COLS)

typedef float v2f __attribute__((ext_vector_type(2)));
typedef float v8f __attribute__((ext_vector_type(8)));

// workspace layout (floats)
#define WS_SUMS   0                  // [C][D]
#define WS_SUMSQ  (C * D)            // [C][D]
#define WS_CNT    (2 * C * D)        // [C]
#define WS_CENT   (2 * C * D + 128)  // [CP][D] centroids (padded rows zero)

__global__ __launch_bounds__(256) void zero_ws(float* __restrict__ ws, int n) {
    int i = blockIdx.x * 256 + threadIdx.x;
    if (i < n) ws[i] = 0.0f;
}

// One-pass segment reduction: per-class sum / sumsq / count into LDS,
// then one atomic flush per (class, column) to global scratch.
__global__ __launch_bounds__(256) void seg_accum(const float* __restrict__ feat,
                                                 const int* __restrict__ labels,
                                                 float* __restrict__ ws, int N) {
    __shared__ float s_sum[C * COLS];
    __shared__ float s_sq[C * COLS];
    __shared__ float s_cnt[C];

    const int tid = threadIdx.x;
    for (int i = tid; i < C * COLS; i += 256) { s_sum[i] = 0.0f; s_sq[i] = 0.0f; }
    if (tid < C) s_cnt[tid] = 0.0f;
    __syncthreads();

    const int wave = tid >> 5;
    const int lane = tid & 31;
    const int cb   = blockIdx.y * COLS;

    const long long rpb = (N + (int)gridDim.x - 1) / (int)gridDim.x;
    const long long r0  = (long long)blockIdx.x * rpb;
    long long r1  = r0 + rpb;
    if (r1 > N) r1 = N;
    const bool countit = (blockIdx.y == 0) && (lane == 0);

    for (long long r = r0 + wave; r < r1; r += 8) {
        const int cls = labels[r];
        v2f v = *(const v2f*)(feat + r * (long long)D + cb + lane * 2);
        float* ss = s_sum + cls * COLS + lane * 2;
        float* sq = s_sq  + cls * COLS + lane * 2;
        __hip_atomic_fetch_add(ss + 0, v.x,       __ATOMIC_RELAXED, __HIP_MEMORY_SCOPE_WORKGROUP);
        __hip_atomic_fetch_add(ss + 1, v.y,       __ATOMIC_RELAXED, __HIP_MEMORY_SCOPE_WORKGROUP);
        __hip_atomic_fetch_add(sq + 0, v.x * v.x, __ATOMIC_RELAXED, __HIP_MEMORY_SCOPE_WORKGROUP);
        __hip_atomic_fetch_add(sq + 1, v.y * v.y, __ATOMIC_RELAXED, __HIP_MEMORY_SCOPE_WORKGROUP);
        if (countit)
            __hip_atomic_fetch_add(&s_cnt[cls], 1.0f, __ATOMIC_RELAXED, __HIP_MEMORY_SCOPE_WORKGROUP);
    }
    __syncthreads();

    for (int i = tid; i < C * COLS; i += 256) {
        const int c   = i / COLS;
        const int col = cb + (i % COLS);
        __hip_atomic_fetch_add(ws + WS_SUMS  + c * D + col, s_sum[i],
                               __ATOMIC_RELAXED, __HIP_MEMORY_SCOPE_AGENT);
        __hip_atomic_fetch_add(ws + WS_SUMSQ + c * D + col, s_sq[i],
                               __ATOMIC_RELAXED, __HIP_MEMORY_SCOPE_AGENT);
    }
    if (blockIdx.y == 0 && tid < C)
        __hip_atomic_fetch_add(ws + WS_CNT + tid, s_cnt[tid],
                               __ATOMIC_RELAXED, __HIP_MEMORY_SCOPE_AGENT);
}

// Single-workgroup epilogue: centroids, per-class variance sum, WMMA Gram
// matrix (fp32-exact V_WMMA_F32_16X16X4_F32), argmax + min-distance + output.
__global__ __launch_bounds__(256) void finalize(float* __restrict__ ws,
                                                float* __restrict__ out) {
    __shared__ float gram[CP * CP];   // 50,176 B
    __shared__ float s_var[CP];
    __shared__ float s_inv[CP];

    const int tid = threadIdx.x;
    if (tid < CP) s_inv[tid] = (tid < C) ? (1.0f / ws[WS_CNT + tid]) : 0.0f;
    __syncthreads();

    // centroids (padded rows = 0) to global scratch
    for (int i = tid; i < CP * D; i += 256) {
        const int c = i / D, d = i % D;
        float cen = (c < C) ? ws[WS_SUMS + c * D + d] * s_inv[c] : 0.0f;
        ws[WS_CENT + i] = cen;
    }
    // per-class within-class variance sum
    if (tid < CP) {
        float acc = 0.0f;
        if (tid < C) {
            const float inv = s_inv[tid];
            for (int d = 0; d < D; ++d) {
                const float s = ws[WS_SUMS  + tid * D + d];
                const float q = ws[WS_SUMSQ + tid * D + d];
                const float m = s * inv;
                acc += q * inv - m * m;
            }
        }
        s_var[tid] = acc;
    }
    __threadfence();
    __syncthreads();

    // Gram = Cent * Cent^T via V_WMMA_F32_16X16X4_F32.
    // ISA 32-bit 16x4 A layout: lanes 0-15 hold K={0,1}, lanes 16-31 hold
    // K={2,3} -> each lane loads a contiguous float2 at col k + 2*hi.
    // B = Cent^T has the mirrored layout, so the same load pattern applies
    // with the tile-j row index.
    const float* cent = ws + WS_CENT;
    const int wave = tid >> 5;
    const int lane = tid & 31;
    const int li = lane & 15;
    const int hi = lane >> 4;
    for (int tile = wave; tile < 49; tile += 8) {
        const int ti = tile / 7, tj = tile % 7;
        const int rowA = ti * 16 + li;
        const int rowB = tj * 16 + li;
        v8f acc = {};
        for (int k = 0; k < D; k += 4) {
            const int col = k + 2 * hi;
            v2f a = *(const v2f*)(cent + rowA * D + col);
            v2f b = *(const v2f*)(cent + rowB * D + col);
            acc = __builtin_amdgcn_wmma_f32_16x16x4_f32(false, a, false, b,
                                                        (short)0, acc, false, false);
        }
        // D layout: VGPR v, lanes 0-15 -> M=v, lanes 16-31 -> M=v+8
        for (int v = 0; v < 8; ++v)
            gram[(ti * 16 + v + 8 * hi) * CP + tj * 16 + li] = acc[v];
    }
    __syncthreads();

    if (tid == 0) {
        int m = 0; float best = s_var[0];
        for (int c2 = 1; c2 < C; ++c2)
            if (s_var[c2] > best) { best = s_var[c2]; m = c2; }
        const float gmm = gram[m * CP + m];
        float mind2 = 3.4e38f;
        for (int c2 = 0; c2 < C; ++c2) {
            if (c2 == m) continue;
            float d2 = gram[c2 * CP + c2] + gmm - 2.0f * gram[m * CP + c2];
            d2 = fmaxf(d2, 0.0f);
            if (d2 < mind2) mind2 = d2;
        }
        out[0] = best / mind2;
    }
}

extern "C" void kernel_launch(void* const* d_in, const int* in_sizes, int n_in,
                              void* d_out, int out_size, void* d_ws, size_t ws_size,
                              hipStream_t stream) {
    const float* feat   = (const float*)d_in[0];
    const int*   labels = (const int*)d_in[1];
    float* out = (float*)d_out;
    float* ws  = (float*)d_ws;
    const int N = in_sizes[1];   // labels count; features are N x 256

    const int zn = 2 * C * D + 128;  // sums + sumsq + counts (+pad)
    zero_ws<<<(zn + 255) / 256, 256, 0, stream>>>(ws, zn);

    dim3 grid(ROW_GROUPS, D / COLS);
    seg_accum<<<grid, 256, 0, stream>>>(feat, labels, ws, N);

    finalize<<<1, 256, 0, stream>>>(ws, out);
}